// ResnetBBlock_42941083025906
// MI455X (gfx1250) — compile-verified
//
#include <hip/hip_runtime.h>

// ---------------------------------------------------------------------------
// KPConv ResNet bottleneck block for MI455X (gfx1250), fp32 WMMA 16x16x4 path.
// ---------------------------------------------------------------------------

typedef float v2f __attribute__((ext_vector_type(2)));
typedef float v8f __attribute__((ext_vector_type(8)));

#define N_PTS   400000
#define M_Q     100000
#define NB      16
#define C_IN    64
#define D2      32
#define C_OUT   128
#define KP      15
#define INV_EXTENT 2.0f   // 1 / (0.5 * 1.0)
#define EPS_BN  1e-5f
#define SLOPE   0.2f

__device__ __forceinline__ float leaky(float v) { return v >= 0.f ? v : SLOPE * v; }

// D = A(16x4) * B(4x16) + C  (fp32 WMMA, CDNA5)
__device__ __forceinline__ v8f wmma4(v2f a, v2f b, v8f c) {
  return __builtin_amdgcn_wmma_f32_16x16x4_f32(false, a, false, b, (short)0, c,
                                               false, false);
}

// A fragment (16x4) from row-major src[ld], rows m0..m0+15, cols k0..k0+3.
// Lane L: row = L&15, holds K = 2*(L>>4) + {0,1}.
__device__ __forceinline__ v2f load_a_rm(const float* src, int ld, int m0,
                                         int k0, int lane) {
  int m = m0 + (lane & 15);
  int k = k0 + ((lane >> 4) << 1);
  v2f a;
  a.x = src[(size_t)m * ld + k];
  a.y = src[(size_t)m * ld + k + 1];
  return a;
}

// B fragment (4x16) from row-major src[ld], rows k0..k0+3, cols n0..n0+15.
__device__ __forceinline__ v2f load_b_rm(const float* src, int ld, int k0,
                                         int n0, int lane) {
  int n = n0 + (lane & 15);
  int k = k0 + ((lane >> 4) << 1);
  v2f b;
  b.x = src[(size_t)k * ld + n];
  b.y = src[(size_t)(k + 1) * ld + n];
  return b;
}

// D tile (16x16): lane L holds col n0+(L&15), rows 8*(L>>4)+v, v=0..7.
__device__ __forceinline__ void store_d_rm(float* dst, int ld, int m0, int n0,
                                           int lane, v8f c) {
  int n = n0 + (lane & 15);
  int rb = m0 + ((lane >> 4) << 3);
#pragma unroll
  for (int v = 0; v < 8; ++v) dst[(size_t)(rb + v) * ld + n] = c[v];
}

__device__ __forceinline__ void accum_stats(float* sum, float* sq, int n0,
                                            int lane, v8f c) {
  float s = 0.f, q = 0.f;
#pragma unroll
  for (int v = 0; v < 8; ++v) { s += c[v]; q += c[v] * c[v]; }
  atomicAdd(&sum[n0 + (lane & 15)], s);
  atomicAdd(&sq[n0 + (lane & 15)], q);
}

// ---------------------------------------------------------------------------
__global__ void k_zero(float* p, int n) {
  int i = blockIdx.x * blockDim.x + threadIdx.x;
  if (i < n) p[i] = 0.f;
}

// a = g*rsqrt(var+eps), c = b - mu*a   (biased variance, matching jnp .var(0))
__global__ void k_finalize_bn(const float* __restrict__ sum,
                              const float* __restrict__ sq,
                              const float* __restrict__ g,
                              const float* __restrict__ b, float cnt,
                              float* a, float* c, int C) {
  int i = blockIdx.x * blockDim.x + threadIdx.x;
  if (i >= C) return;
  float mu  = sum[i] / cnt;
  float var = sq[i] / cnt - mu * mu;
  float s   = g[i] * rsqrtf(fmaxf(var, 0.f) + EPS_BN);
  a[i] = s;
  c[i] = b[i] - mu * s;
}

// ---------------------------------------------------------------------------
// y1 = x[N,64] @ Wu1[64,32]; per-column sum/sumsq for BN1.
__global__ __launch_bounds__(256) void k_gemm1(const float* __restrict__ x,
                                               const float* __restrict__ W,
                                               float* __restrict__ y1,
                                               float* sum1, float* sq1,
                                               int ntiles) {
  int lane = threadIdx.x & 31;
  int tile = blockIdx.x * (blockDim.x >> 5) + (threadIdx.x >> 5);
  if (tile >= ntiles) return;
  int m0 = tile * 16;
  v8f acc0 = {}, acc1 = {};
#pragma unroll
  for (int k0 = 0; k0 < C_IN; k0 += 4) {
    v2f a  = load_a_rm(x, C_IN, m0, k0, lane);
    v2f b0 = load_b_rm(W, D2, k0, 0, lane);
    v2f b1 = load_b_rm(W, D2, k0, 16, lane);
    acc0 = wmma4(a, b0, acc0);
    acc1 = wmma4(a, b1, acc1);
  }
  store_d_rm(y1, D2, m0, 0, lane, acc0);
  store_d_rm(y1, D2, m0, 16, lane, acc1);
  accum_stats(sum1, sq1, 0, lane, acc0);
  accum_stats(sum1, sq1, 16, lane, acc1);
}

// ---------------------------------------------------------------------------
// KPConv core: 16 queries per block, 128 threads (4 waves).
//   phase0: build w[16q][16nb][16kp(pad)] and bn+act'd nf[16q][16nb][32d] in LDS
//   stepA : per query, weighted[16kp x 32d] = w^T @ nf  (WMMA, regs)
//   stepB : y2[16q][32] = weighted_flat[16q][512pad] @ KV[512pad][32] (WMMA,
//           split-K over 4 waves + LDS reduction); stats atomics.
__global__ __launch_bounds__(128) void k_kpconv(
    const float* __restrict__ y1, const int* __restrict__ idx,
    const float* __restrict__ pos, const float* __restrict__ qpos,
    const float* __restrict__ Kpts, const float* __restrict__ KV,
    const float* __restrict__ a1, const float* __restrict__ c1,
    float* __restrict__ y2, float* sumk, float* sqk) {
  __shared__ float smem[12288];          // 48 KB
  float* s_w  = smem;                    // [16][16][16] (q, nb, kp) kp15 = 0
  float* s_nf = smem + 4096;             // [16][16][32] (q, nb, d)
  int lane = threadIdx.x & 31;
  int wave = threadIdx.x >> 5;
  int q0 = blockIdx.x * 16;

  // ---- phase 0
  for (int p = threadIdx.x; p < 256; p += 128) {
    int q = p >> 4, nb = p & 15;
    int id = idx[(size_t)(q0 + q) * NB + nb];
    float qx = qpos[(size_t)(q0 + q) * 3 + 0];
    float qy = qpos[(size_t)(q0 + q) * 3 + 1];
    float qz = qpos[(size_t)(q0 + q) * 3 + 2];
    float px, py, pz;
    if (id < N_PTS) {
      px = pos[(size_t)id * 3 + 0];
      py = pos[(size_t)id * 3 + 1];
      pz = pos[(size_t)id * 3 + 2];
    } else {
      px = py = pz = 1e6f;               // shadow support point
    }
    float rx = px - qx, ry = py - qy, rz = pz - qz;
    float rr = rx * rx + ry * ry + rz * rz;
#pragma unroll
    for (int k = 0; k < KP; ++k) {
      float kx = Kpts[k * 3 + 0], ky = Kpts[k * 3 + 1], kz = Kpts[k * 3 + 2];
      float sq = rr + kx * kx + ky * ky + kz * kz
                 - 2.f * (rx * kx + ry * ky + rz * kz);
      float w = 1.f - sqrtf(fmaxf(sq, 0.f)) * INV_EXTENT;
      s_w[q * 256 + nb * 16 + k] = fmaxf(w, 0.f);
    }
    s_w[q * 256 + nb * 16 + 15] = 0.f;
    if (id < N_PTS) {
      const float* yr = y1 + (size_t)id * D2;
#pragma unroll
      for (int d = 0; d < D2; ++d)
        s_nf[q * 512 + nb * 32 + d] = leaky(yr[d] * a1[d] + c1[d]);
    } else {
#pragma unroll
      for (int d = 0; d < D2; ++d) s_nf[q * 512 + nb * 32 + d] = 0.f;
    }
  }
  __syncthreads();

  // ---- step A: wave handles queries wave*4 .. +3
  v8f wt[4][2];
#pragma unroll
  for (int qq = 0; qq < 4; ++qq) { wt[qq][0] = (v8f){}; wt[qq][1] = (v8f){}; }
#pragma unroll
  for (int qq = 0; qq < 4; ++qq) {
    int q = wave * 4 + qq;
#pragma unroll
    for (int k0 = 0; k0 < 16; k0 += 4) {
      int m = lane & 15;
      int kk = k0 + ((lane >> 4) << 1);
      v2f a;                              // A[kp][nb] = s_w[q][nb][kp]
      a.x = s_w[q * 256 + kk * 16 + m];
      a.y = s_w[q * 256 + (kk + 1) * 16 + m];
#pragma unroll
      for (int t = 0; t < 2; ++t) {
        int n = t * 16 + (lane & 15);
        v2f b;                            // B[nb][d] = s_nf[q][nb][d]
        b.x = s_nf[q * 512 + kk * 32 + n];
        b.y = s_nf[q * 512 + (kk + 1) * 32 + n];
        wt[qq][t] = wmma4(a, b, wt[qq][t]);
      }
    }
  }
  __syncthreads();

  // overlay weighted_flat [16q][512] on top of s_w/s_nf
  float* s_wt = smem;
#pragma unroll
  for (int qq = 0; qq < 4; ++qq) {
    int q = wave * 4 + qq;
#pragma unroll
    for (int t = 0; t < 2; ++t) {
      int n = t * 16 + (lane & 15);
      int rb = (lane >> 4) << 3;
#pragma unroll
      for (int v = 0; v < 8; ++v)
        s_wt[q * 512 + (rb + v) * 32 + n] = wt[qq][t][v];
    }
  }
  __syncthreads();

  // ---- step B: split-K: wave -> ntile = wave&1, khalf = wave>>1
  int ntile = wave & 1;
  int kbase = (wave >> 1) * 256;
  v8f acc = {};
  for (int k0 = 0; k0 < 256; k0 += 4) {
    int kk = kbase + k0 + ((lane >> 4) << 1);
    int m = lane & 15;
    v2f a;
    a.x = s_wt[m * 512 + kk];
    a.y = s_wt[m * 512 + kk + 1];
    int n = ntile * 16 + (lane & 15);
    v2f b;                                // KV flat [480][32], rows>=480 -> 0
    b.x = (kk < 480) ? KV[(size_t)kk * 32 + n] : 0.f;
    b.y = (kk + 1 < 480) ? KV[(size_t)(kk + 1) * 32 + n] : 0.f;
    acc = wmma4(a, b, acc);
  }
  __syncthreads();
  float* s_red = smem;                    // 2 tiles * 256 floats
  if (wave >= 2) {
#pragma unroll
    for (int v = 0; v < 8; ++v) s_red[(wave - 2) * 256 + lane * 8 + v] = acc[v];
  }
  __syncthreads();
  if (wave < 2) {
#pragma unroll
    for (int v = 0; v < 8; ++v) acc[v] += s_red[wave * 256 + lane * 8 + v];
    int col = wave * 16 + (lane & 15);
    int rb = q0 + ((lane >> 4) << 3);
    float s = 0.f, q2 = 0.f;
#pragma unroll
    for (int v = 0; v < 8; ++v) {
      float val = acc[v];
      y2[(size_t)(rb + v) * D2 + col] = val;
      s += val;
      q2 += val * val;
    }
    atomicAdd(&sumk[col], s);
    atomicAdd(&sqk[col], q2);
  }
}

// ---------------------------------------------------------------------------
// y3 = act(bnK(y2)) @ Wu2[32,128]; stats for BN2.
__global__ __launch_bounds__(256) void k_gemm2(const float* __restrict__ y2,
                                               const float* __restrict__ W,
                                               const float* __restrict__ ak,
                                               const float* __restrict__ ck,
                                               float* __restrict__ y3,
                                               float* sum2, float* sq2,
                                               int ntiles) {
  int lane = threadIdx.x & 31;
  int tile = blockIdx.x * (blockDim.x >> 5) + (threadIdx.x >> 5);
  if (tile >= ntiles) return;
  int m0 = tile * 16;
  v8f acc[8];
#pragma unroll
  for (int t = 0; t < 8; ++t) acc[t] = (v8f){};
#pragma unroll
  for (int k0 = 0; k0 < D2; k0 += 4) {
    int m = m0 + (lane & 15);
    int kk = k0 + ((lane >> 4) << 1);
    v2f a;                                // bn + leaky on the fly
    a.x = leaky(y2[(size_t)m * D2 + kk] * ak[kk] + ck[kk]);
    a.y = leaky(y2[(size_t)m * D2 + kk + 1] * ak[kk + 1] + ck[kk + 1]);
#pragma unroll
    for (int t = 0; t < 8; ++t) {
      v2f b = load_b_rm(W, C_OUT, k0, t * 16, lane);
      acc[t] = wmma4(a, b, acc[t]);
    }
  }
#pragma unroll
  for (int t = 0; t < 8; ++t) {
    store_d_rm(y3, C_OUT, m0, t * 16, lane, acc[t]);
    accum_stats(sum2, sq2, t * 16, lane, acc[t]);
  }
}

// ---------------------------------------------------------------------------
// Shortcut (neighbor max-pool of x, zero shadow row) @ Wsc, plus act(bn2(y3)).
__global__ __launch_bounds__(128) void k_final(const float* __restrict__ x,
                                               const int* __restrict__ idx,
                                               const float* __restrict__ Wsc,
                                               const float* __restrict__ y3,
                                               const float* __restrict__ a2,
                                               const float* __restrict__ c2,
                                               float* __restrict__ out) {
  __shared__ float s_sx[16 * 64];
  int q0 = blockIdx.x * 16;
  for (int p = threadIdx.x; p < 1024; p += 128) {
    int q = p >> 6, c = p & 63;
    float acc = -3.4e38f;
#pragma unroll
    for (int nb = 0; nb < NB; ++nb) {
      int id = idx[(size_t)(q0 + q) * NB + nb];
      float v = (id < N_PTS) ? x[(size_t)id * C_IN + c] : 0.f;
      acc = fmaxf(acc, v);
    }
    s_sx[q * 64 + c] = acc;
  }
  __syncthreads();
  int lane = threadIdx.x & 31;
  int wave = threadIdx.x >> 5;
#pragma unroll
  for (int tt = 0; tt < 2; ++tt) {
    int ntile = wave * 2 + tt;
    v8f acc = {};
#pragma unroll
    for (int k0 = 0; k0 < C_IN; k0 += 4) {
      int m = lane & 15;
      int kk = k0 + ((lane >> 4) << 1);
      v2f a;
      a.x = s_sx[m * 64 + kk];
      a.y = s_sx[m * 64 + kk + 1];
      v2f b = load_b_rm(Wsc, C_OUT, k0, ntile * 16, lane);
      acc = wmma4(a, b, acc);
    }
    int col = ntile * 16 + (lane & 15);
    int rb = (lane >> 4) << 3;
#pragma unroll
    for (int v = 0; v < 8; ++v) {
      int m = q0 + rb + v;
      float h = leaky(y3[(size_t)m * C_OUT + col] * a2[col] + c2[col]);
      out[(size_t)m * C_OUT + col] = h + acc[v];
    }
  }
}

// ---------------------------------------------------------------------------
extern "C" void kernel_launch(void* const* d_in, const int* in_sizes, int n_in,
                              void* d_out, int out_size, void* d_ws,
                              size_t ws_size, hipStream_t stream) {
  const float* x    = (const float*)d_in[0];
  const float* pos  = (const float*)d_in[1];
  const float* qpos = (const float*)d_in[2];
  const int*   idx  = (const int*)d_in[3];
  const float* Wu1  = (const float*)d_in[4];
  const float* g1   = (const float*)d_in[5];
  const float* b1   = (const float*)d_in[6];
  const float* Kpts = (const float*)d_in[7];
  const float* KV   = (const float*)d_in[8];
  const float* gk   = (const float*)d_in[9];
  const float* bk   = (const float*)d_in[10];
  const float* Wu2  = (const float*)d_in[11];
  const float* g2   = (const float*)d_in[12];
  const float* b2   = (const float*)d_in[13];
  const float* Wsc  = (const float*)d_in[14];
  float* out = (float*)d_out;

  float* ws = (float*)d_ws;
  float* y1 = ws;                                   // [N,32]
  float* y2 = y1 + (size_t)N_PTS * D2;              // [M,32]
  float* y3 = y2 + (size_t)M_Q * D2;                // [M,128]
  float* S  = y3 + (size_t)M_Q * C_OUT;             // stats block
  float *sum1 = S +   0, *sq1 = S +  32;
  float *sumk = S +  64, *sqk = S +  96;
  float *sum2 = S + 128, *sq2 = S + 256;
  float *a1 = S + 384, *c1 = S + 416;
  float *ak = S + 448, *ck = S + 480;
  float *a2 = S + 512, *c2 = S + 640;

  k_zero<<<1, 384, 0, stream>>>(S, 384);

  // unary_1 + BN1 stats
  k_gemm1<<<(N_PTS / 16) / 8, 256, 0, stream>>>(x, Wu1, y1, sum1, sq1,
                                                N_PTS / 16);
  k_finalize_bn<<<1, 32, 0, stream>>>(sum1, sq1, g1, b1, (float)N_PTS, a1, c1,
                                      D2);
  // KPConv + BNk stats
  k_kpconv<<<M_Q / 16, 128, 0, stream>>>(y1, idx, pos, qpos, Kpts, KV, a1, c1,
                                         y2, sumk, sqk);
  k_finalize_bn<<<1, 32, 0, stream>>>(sumk, sqk, gk, bk, (float)M_Q, ak, ck,
                                      D2);
  // unary_2 + BN2 stats
  int tiles2 = M_Q / 16;
  k_gemm2<<<(tiles2 + 7) / 8, 256, 0, stream>>>(y2, Wu2, ak, ck, y3, sum2, sq2,
                                                tiles2);
  k_finalize_bn<<<1, 128, 0, stream>>>(sum2, sq2, g2, b2, (float)M_Q, a2, c2,
                                       C_OUT);
  // shortcut + combine
  k_final<<<M_Q / 16, 128, 0, stream>>>(x, idx, Wsc, y3, a2, c2, out);
}